// LFA_72464688218272
// MI455X (gfx1250) — compile-verified
//
#include <hip/hip_runtime.h>

typedef __attribute__((ext_vector_type(2))) float v2f;
typedef __attribute__((ext_vector_type(8))) float v8f;

// ---------------------------------------------------------------------------
// Pointwise conv (GEMM) + folded BN + ReLU using V_WMMA_F32_16X16X4_F32.
// A = weight tile (M=out-channels, K=in-channels), B = activations
// (K=in-channels, N=points), D[ch][pt]. Exact fp32 path (workload is
// memory-bound: ~11 GFLOP vs ~0.5 GB traffic, so no reason to drop precision).
// A-frag (16x4 f32): lanes 0-15 hold {K=kk,kk+1}, lanes 16-31 {K=kk+2,kk+3};
// B-frag mirrors the K split; D: lane<16,vgpr r -> D[r][lane],
// lane>=16 -> D[r+8][lane-16]  => per lane: 8 consecutive channels of one
// point, stored with two b128 stores (point-major out).
// CIN and the input-layout flag are compile-time so the inner loop is pure
// load+wmma (no per-iteration scalar branching).
// ---------------------------------------------------------------------------
template <int CIN, bool IN_CHAN_MAJOR>
__global__ __launch_bounds__(256) void cbr_gemm_kernel(
    const float* __restrict__ in, const float* __restrict__ W,
    const float* __restrict__ s, const float* __restrict__ bias,
    float* __restrict__ out, int B, int N, int Cout)
{
  int wid = (blockIdx.x * blockDim.x + threadIdx.x) >> 5;
  int ctiles = Cout >> 4;
  int ntiles = N >> 4;
  int total = B * ntiles * ctiles;
  if (wid >= total) return;                 // wave-uniform (EXEC all-1 for WMMA)
  int ct = wid % ctiles;
  int rest = wid / ctiles;
  int b = rest / ntiles;
  int n0 = (rest % ntiles) << 4;
  int c0 = ct << 4;

  int lane = threadIdx.x & 31;
  int hi = lane >> 4;                       // 0 or 1
  int l = lane & 15;
  int arow = c0 + l;                        // output channel for A fragment
  int kofs = hi ? 2 : 0;
  int pt = n0 + l;

  const float* wrow = W + (long)arow * CIN + kofs;
  const float* brow = IN_CHAN_MAJOR
      ? in + (long)b * CIN * N + (long)kofs * N + pt      // in[b][c][n]
      : in + ((long)(b * N + pt)) * CIN + kofs;           // in[b*N+n][c]

  v8f acc = {};
#pragma unroll
  for (int kk = 0; kk < CIN; kk += 4) {
    v2f a, bf;
    a.x = wrow[kk];
    a.y = wrow[kk + 1];
    if (IN_CHAN_MAJOR) {
      bf.x = brow[(long)kk * N];
      bf.y = brow[(long)kk * N + N];
    } else {
      const float2 t = *(const float2*)(brow + kk);
      bf.x = t.x; bf.y = t.y;
    }
    acc = __builtin_amdgcn_wmma_f32_16x16x4_f32(false, a, false, bf, (short)0, acc,
                                                false, false);
  }

  // Epilogue: lane holds channels [c0+hi*8, c0+hi*8+8) of point pt.
  int cb = c0 + hi * 8;
  const float4 sa = *(const float4*)(s + cb);
  const float4 sb = *(const float4*)(s + cb + 4);
  const float4 ba = *(const float4*)(bias + cb);
  const float4 bb = *(const float4*)(bias + cb + 4);
  float4 o0, o1;
  o0.x = fmaxf(fmaf(sa.x, acc[0], ba.x), 0.0f);
  o0.y = fmaxf(fmaf(sa.y, acc[1], ba.y), 0.0f);
  o0.z = fmaxf(fmaf(sa.z, acc[2], ba.z), 0.0f);
  o0.w = fmaxf(fmaf(sa.w, acc[3], ba.w), 0.0f);
  o1.x = fmaxf(fmaf(sb.x, acc[4], bb.x), 0.0f);
  o1.y = fmaxf(fmaf(sb.y, acc[5], bb.y), 0.0f);
  o1.z = fmaxf(fmaf(sb.z, acc[6], bb.z), 0.0f);
  o1.w = fmaxf(fmaf(sb.w, acc[7], bb.w), 0.0f);
  float* op = out + ((long)(b * N + pt)) * Cout + cb;
  *(float4*)(op) = o0;
  *(float4*)(op + 4) = o1;
}

// ---------------------------------------------------------------------------
// Neighborhood aggregation (gather-bound stage): one wave per point.
// agg[p][c] = mean_j ReLU(sx[c]*(Wx[c]·rel_j)+bx[c]) * feats[idx_j][c], c<64.
// Lanes 0-15 fetch neighbor idx/xyz; rel broadcast with __shfl (wave32).
// Each lane owns channels {2*lane, 2*lane+1} -> coalesced 256B feature rows.
// ---------------------------------------------------------------------------
__global__ __launch_bounds__(256) void lfa_aggregate_kernel(
    const float* __restrict__ xyz, const int* __restrict__ idx,
    const float* __restrict__ feats,        // [B*N, 64] point-major
    const float* __restrict__ Wx,           // [>=64, 3] row-major (rows 0..63 used)
    const float* __restrict__ sx, const float* __restrict__ bx,
    float* __restrict__ agg,                // [B*N, 64]
    int B, int N)
{
  const int K = 16;
  int p = (blockIdx.x * blockDim.x + threadIdx.x) >> 5;
  if (p >= B * N) return;
  int lane = threadIdx.x & 31;
  int b = p / N;

  float cx = xyz[(long)p * 3 + 0];
  float cy = xyz[(long)p * 3 + 1];
  float cz = xyz[(long)p * 3 + 2];

  int j = lane & 15;
  int nb = idx[((long)p) * K + j];          // neighbor index within batch
  long nbase = ((long)(b * N + nb)) * 3;
  float rx = cx - xyz[nbase + 0];
  float ry = cy - xyz[nbase + 1];
  float rz = cz - xyz[nbase + 2];

  int c0 = lane * 2;
  float w00 = Wx[c0 * 3 + 0], w01 = Wx[c0 * 3 + 1], w02 = Wx[c0 * 3 + 2];
  float w10 = Wx[(c0 + 1) * 3 + 0], w11 = Wx[(c0 + 1) * 3 + 1], w12 = Wx[(c0 + 1) * 3 + 2];
  float s0 = sx[c0], s1 = sx[c0 + 1];
  float bb0 = bx[c0], bb1 = bx[c0 + 1];

  const float* fb = feats + ((long)b * N) * 64 + c0;
  float acc0 = 0.0f, acc1 = 0.0f;
#pragma unroll 4
  for (int jj = 0; jj < K; ++jj) {
    int nbj = __shfl(nb, jj, 32);
    float rxx = __shfl(rx, jj, 32);
    float ryy = __shfl(ry, jj, 32);
    float rzz = __shfl(rz, jj, 32);
    float t0 = fmaf(w00, rxx, fmaf(w01, ryy, w02 * rzz));
    float t1 = fmaf(w10, rxx, fmaf(w11, ryy, w12 * rzz));
    t0 = fmaxf(fmaf(s0, t0, bb0), 0.0f);
    t1 = fmaxf(fmaf(s1, t1, bb1), 0.0f);
    const float2 fn = *(const float2*)(fb + (long)nbj * 64);
    acc0 = fmaf(t0, fn.x, acc0);
    acc1 = fmaf(t1, fn.y, acc1);
  }
  float2 o;
  o.x = acc0 * (1.0f / 16.0f);
  o.y = acc1 * (1.0f / 16.0f);
  *(float2*)(agg + (long)p * 64 + c0) = o;
}

// ---------------------------------------------------------------------------
// Final fused stage: main branch (wm2: 256x128 over f3, point-major) +
// shortcut branch (wsc: 256x32 over feature, channel-major), both on WMMA,
// per-branch BN+ReLU, sum, leaky ReLU(0.2), channel-major coalesced store.
// ---------------------------------------------------------------------------
__global__ __launch_bounds__(256) void lfa_final_kernel(
    const float* __restrict__ f3,           // [B*N, 128] point-major
    const float* __restrict__ feature,      // [B, 32, N] channel-major
    const float* __restrict__ Wm2, const float* __restrict__ sm2, const float* __restrict__ bm2,
    const float* __restrict__ Wsc, const float* __restrict__ ssc, const float* __restrict__ bsc,
    float* __restrict__ out, int B, int N)
{
  const int Cout = 256, Cm = 128, Cs = 32;
  int wid = (blockIdx.x * blockDim.x + threadIdx.x) >> 5;
  int ctiles = Cout >> 4;
  int ntiles = N >> 4;
  int total = B * ntiles * ctiles;
  if (wid >= total) return;
  int ct = wid % ctiles;
  int rest = wid / ctiles;
  int b = rest / ntiles;
  int n0 = (rest % ntiles) << 4;
  int c0 = ct << 4;

  int lane = threadIdx.x & 31;
  int hi = lane >> 4;
  int l = lane & 15;
  int arow = c0 + l;
  int kofs = hi ? 2 : 0;
  int pt = n0 + l;

  const float* wm = Wm2 + (long)arow * Cm + kofs;
  const float* bm = f3 + ((long)(b * N + pt)) * Cm + kofs;
  v8f am = {};
#pragma unroll 8
  for (int kk = 0; kk < Cm; kk += 4) {      // main branch: f3 @ wm2^T
    v2f a, bf;
    a.x = wm[kk];
    a.y = wm[kk + 1];
    const float2 t = *(const float2*)(bm + kk);
    bf.x = t.x; bf.y = t.y;
    am = __builtin_amdgcn_wmma_f32_16x16x4_f32(false, a, false, bf, (short)0, am,
                                               false, false);
  }

  const float* ws = Wsc + (long)arow * Cs + kofs;
  const float* fb = feature + (long)b * Cs * N + (long)kofs * N + pt;
  v8f as = {};
#pragma unroll
  for (int kk = 0; kk < Cs; kk += 4) {      // shortcut: feature @ wsc^T
    v2f a, bf;
    a.x = ws[kk];
    a.y = ws[kk + 1];
    bf.x = fb[(long)kk * N];
    bf.y = fb[(long)kk * N + N];
    as = __builtin_amdgcn_wmma_f32_16x16x4_f32(false, a, false, bf, (short)0, as,
                                               false, false);
  }

#pragma unroll
  for (int r = 0; r < 8; ++r) {
    int ch = c0 + r + (hi ? 8 : 0);
    float m = fmaxf(fmaf(sm2[ch], am[r], bm2[ch]), 0.0f);
    float sc = fmaxf(fmaf(ssc[ch], as[r], bsc[ch]), 0.0f);
    float o = m + sc;
    o = (o >= 0.0f) ? o : 0.2f * o;         // leaky_relu (identity here, kept for fidelity)
    out[(long)b * Cout * N + (long)ch * N + n0 + l] = o;
  }
}

// ---------------------------------------------------------------------------
extern "C" void kernel_launch(void* const* d_in, const int* in_sizes, int n_in,
                              void* d_out, int out_size, void* d_ws, size_t ws_size,
                              hipStream_t stream) {
  const float* feature = (const float*)d_in[0];   // [B,32,N,1]
  const float* xyz     = (const float*)d_in[1];   // [B,N,3]
  const int*   nidx    = (const int*)d_in[2];     // [B,N,16]
  const float* wm1 = (const float*)d_in[3];
  const float* sm1 = (const float*)d_in[4];
  const float* bm1 = (const float*)d_in[5];
  const float* w1x = (const float*)d_in[6];
  const float* s1x = (const float*)d_in[7];
  const float* b1x = (const float*)d_in[8];
  const float* w1f = (const float*)d_in[9];
  const float* s1f = (const float*)d_in[10];
  const float* b1f = (const float*)d_in[11];
  const float* w2x = (const float*)d_in[12];
  const float* s2x = (const float*)d_in[13];
  const float* b2x = (const float*)d_in[14];
  const float* w2f = (const float*)d_in[15];
  const float* s2f = (const float*)d_in[16];
  const float* b2f = (const float*)d_in[17];
  const float* wm2 = (const float*)d_in[18];
  const float* sm2 = (const float*)d_in[19];
  const float* bm2 = (const float*)d_in[20];
  const float* wsc = (const float*)d_in[21];
  const float* ssc = (const float*)d_in[22];
  const float* bsc = (const float*)d_in[23];
  float* out = (float*)d_out;

  const int B = 2;
  const int N = (in_sizes[1] / 3) / B;            // 40960
  const long P = (long)B * N;

  // Workspace carve-up (agg2 aliases f1, which is dead after block1 gather).
  float* f1   = (float*)d_ws;                     // P*64
  float* agg1 = f1 + P * 64;                      // P*64
  float* f2   = agg1 + P * 64;                    // P*64
  float* f3   = f2 + P * 64;                      // P*128
  float* agg2 = f1;

  const int THREADS = 256;
  const int WPB = THREADS / 32;

  // 1) f1 = CBR(feature, wm1)   Cin=32 (chan-major in), Cout=64, point-major out
  {
    int waves = B * (N / 16) * (64 / 16);
    cbr_gemm_kernel<32, true><<<(waves + WPB - 1) / WPB, THREADS, 0, stream>>>(
        feature, wm1, sm1, bm1, f1, B, N, 64);
  }
  // 2) agg1 = block1 gather/aggregate
  {
    long waves = P;
    lfa_aggregate_kernel<<<(int)((waves + WPB - 1) / WPB), THREADS, 0, stream>>>(
        xyz, nidx, f1, w1x, s1x, b1x, agg1, B, N);
  }
  // 3) f2 = CBR(agg1, w1f)      Cin=64, Cout=64, point-major both
  {
    int waves = B * (N / 16) * (64 / 16);
    cbr_gemm_kernel<64, false><<<(waves + WPB - 1) / WPB, THREADS, 0, stream>>>(
        agg1, w1f, s1f, b1f, f2, B, N, 64);
  }
  // 4) agg2 = block2 gather/aggregate (first 64 rows of w2x/s2x/b2x)
  {
    long waves = P;
    lfa_aggregate_kernel<<<(int)((waves + WPB - 1) / WPB), THREADS, 0, stream>>>(
        xyz, nidx, f2, w2x, s2x, b2x, agg2, B, N);
  }
  // 5) f3 = CBR(agg2, w2f)      Cin=64, Cout=128, point-major both
  {
    int waves = B * (N / 16) * (128 / 16);
    cbr_gemm_kernel<64, false><<<(waves + WPB - 1) / WPB, THREADS, 0, stream>>>(
        agg2, w2f, s2f, b2f, f3, B, N, 128);
  }
  // 6) out = leaky(CBR(f3, wm2) + CBR(feature, wsc))  Cout=256, chan-major out
  {
    int waves = B * (N / 16) * (256 / 16);
    lfa_final_kernel<<<(waves + WPB - 1) / WPB, THREADS, 0, stream>>>(
        f3, feature, wm2, sm2, bm2, wsc, ssc, bsc, out, B, N);
  }
}